// RADAnomalyHead_34583076667656
// MI455X (gfx1250) — compile-verified
//
#include <hip/hip_runtime.h>
#include <hip/hip_bf16.h>
#include <math.h>

typedef __attribute__((ext_vector_type(2))) float v2f;
typedef __attribute__((ext_vector_type(8))) float v8f;

#define NL 4
#define BATCH 8
#define LPATCH 196
#define DIM 768
#define NBANK 800
#define KTOP 5
#define EPSN 1e-12f
#define KC 32   // K chunk staged in LDS

// ---------------------------------------------------------------------------
// Kernel 1: cls cosine sims (8 x 800) + top-5 per image (tie -> lowest index)
// ---------------------------------------------------------------------------
__global__ __launch_bounds__(256) void k_cls_topk(const float* __restrict__ cls_feats,
                                                  const float* __restrict__ cls_banks,
                                                  int* __restrict__ topk) {
  const int b    = blockIdx.x;
  const int tid  = threadIdx.x;
  const int wave = tid >> 5;
  const int lane = tid & 31;

  __shared__ float sx[DIM];
  __shared__ float srow[NBANK];
  __shared__ float sval[256];
  __shared__ int   sidx[256];
  __shared__ float sxinv;

  const float* x = cls_feats + ((size_t)(NL - 1) * BATCH + b) * DIM;
  for (int i = tid; i < DIM; i += 256) sx[i] = x[i];
  __syncthreads();

  if (wave == 0) {
    float s = 0.f;
    for (int i = lane; i < DIM; i += 32) { float v = sx[i]; s += v * v; }
    #pragma unroll
    for (int m = 16; m >= 1; m >>= 1) s += __shfl_xor(s, m, 32);
    if (lane == 0) sxinv = 1.0f / fmaxf(sqrtf(s), EPSN);
  }
  __syncthreads();
  const float invx = sxinv;

  for (int j = wave; j < NBANK; j += 8) {
    const float* y = cls_banks + ((size_t)(NL - 1) * NBANK + j) * DIM;
    float d = 0.f, q = 0.f;
    for (int i = lane; i < DIM; i += 32) {
      float yv = y[i];
      d += yv * sx[i];
      q += yv * yv;
    }
    #pragma unroll
    for (int m = 16; m >= 1; m >>= 1) {
      d += __shfl_xor(d, m, 32);
      q += __shfl_xor(q, m, 32);
    }
    if (lane == 0) srow[j] = d * invx * (1.0f / fmaxf(sqrtf(q), EPSN));
  }
  __syncthreads();

  for (int r = 0; r < KTOP; ++r) {
    float bv = -INFINITY; int bi = -1;
    for (int j = tid; j < NBANK; j += 256) {
      float v = srow[j];
      if (v > bv) { bv = v; bi = j; }
    }
    sval[tid] = bv; sidx[tid] = bi;
    __syncthreads();
    for (int str = 128; str >= 1; str >>= 1) {
      if (tid < str) {
        float ov = sval[tid + str]; int oi = sidx[tid + str];
        if (ov > sval[tid] || (ov == sval[tid] && oi >= 0 && oi < sidx[tid])) {
          sval[tid] = ov; sidx[tid] = oi;
        }
      }
      __syncthreads();
    }
    if (tid == 0) { topk[b * KTOP + r] = sidx[0]; srow[sidx[0]] = -INFINITY; }
    __syncthreads();
  }
}

// ---------------------------------------------------------------------------
// Kernel 2: inverse L2 norms of query patch rows (wave per row)
// ---------------------------------------------------------------------------
__global__ __launch_bounds__(256) void k_rownorm_q(const float* __restrict__ pf,
                                                   float* __restrict__ qinv, int nrows) {
  const int row  = blockIdx.x * 8 + (threadIdx.x >> 5);
  const int lane = threadIdx.x & 31;
  if (row >= nrows) return;
  const float* p = pf + (size_t)row * DIM;
  float s = 0.f;
  for (int i = lane; i < DIM; i += 32) { float v = p[i]; s += v * v; }
  #pragma unroll
  for (int m = 16; m >= 1; m >>= 1) s += __shfl_xor(s, m, 32);
  if (lane == 0) qinv[row] = 1.0f / fmaxf(sqrtf(s), EPSN);
}

// ---------------------------------------------------------------------------
// Kernel 3: inverse L2 norms of selected bank rows [n][b][k][m]
// ---------------------------------------------------------------------------
__global__ __launch_bounds__(256) void k_rownorm_bank(const float* __restrict__ pb,
                                                      const int* __restrict__ topk,
                                                      float* __restrict__ binv) {
  const int r    = blockIdx.x * 8 + (threadIdx.x >> 5);
  const int lane = threadIdx.x & 31;
  if (r >= NL * BATCH * KTOP * LPATCH) return;
  const int m = r % LPATCH;
  const int k = (r / LPATCH) % KTOP;
  const int b = (r / (LPATCH * KTOP)) % BATCH;
  const int n = r / (LPATCH * KTOP * BATCH);
  const int idx = topk[b * KTOP + k];
  const float* p = pb + (((size_t)n * NBANK + idx) * LPATCH + m) * DIM;
  float s = 0.f;
  for (int i = lane; i < DIM; i += 32) { float v = p[i]; s += v * v; }
  #pragma unroll
  for (int mm = 16; mm >= 1; mm >>= 1) s += __shfl_xor(s, mm, 32);
  if (lane == 0) binv[r] = 1.0f / fmaxf(sqrtf(s), EPSN);
}

// ---------------------------------------------------------------------------
// Kernel 4: WMMA f32 GEMM + row max.  Grid (13 rowtiles, 8 batch, 4 layers).
// sim(l,m) = (a_l . b_m) * invA_l * invB_m ; maxsim[n][b][l] = max over k,m.
// ---------------------------------------------------------------------------
__global__ __launch_bounds__(256) void k_gemm(const float* __restrict__ pf,
                                              const float* __restrict__ pb,
                                              const int* __restrict__ topk,
                                              const float* __restrict__ qinv,
                                              const float* __restrict__ binv,
                                              float* __restrict__ maxsim) {
  const int rowtile = blockIdx.x;   // 0..12
  const int b       = blockIdx.y;   // 0..7
  const int n       = blockIdx.z;   // 0..3
  const int tid  = threadIdx.x;
  const int wave = tid >> 5;
  const int lane = tid & 31;
  const int half = lane >> 4;       // 0: K+0/K+1 , 1: K+2/K+3
  const int l16  = lane & 15;

  __shared__ float ldsA[16][KC + 2];
  __shared__ float ldsB[208][KC + 2];
  __shared__ float swmax[8][16];

  const int rowbase = rowtile * 16;
  const size_t arow0 = (size_t)(n * BATCH + b) * LPATCH;
  const float* Aptr = pf + (arow0 + rowbase) * DIM;

  // column tiles owned by this wave (clamped, never divergent around WMMA)
  const int t0 = wave;                       // 0..7
  const int t1 = wave + 8;                   // 8..15 (valid if < 13)
  const int t1c = (t1 < 13) ? t1 : 12;

  float mx[8];
  #pragma unroll
  for (int i = 0; i < 8; ++i) mx[i] = -INFINITY;

  for (int kb = 0; kb < KTOP; ++kb) {
    const int bankidx = topk[b * KTOP + kb];
    const float* Bptr = pb + ((size_t)n * NBANK + bankidx) * LPATCH * DIM;

    v8f acc0 = {0.f, 0.f, 0.f, 0.f, 0.f, 0.f, 0.f, 0.f};
    v8f acc1 = {0.f, 0.f, 0.f, 0.f, 0.f, 0.f, 0.f, 0.f};

    for (int kc0 = 0; kc0 < DIM; kc0 += KC) {
      __syncthreads();
      // stage A chunk (16 x KC), zero-pad rows >= 196
      for (int idx = tid; idx < 16 * KC; idx += 256) {
        const int r = idx / KC, c = idx % KC;
        float v = 0.f;
        if (rowbase + r < LPATCH) v = Aptr[(size_t)r * DIM + kc0 + c];
        ldsA[r][c] = v;
      }
      // stage B chunk (208 x KC), zero-pad rows >= 196
      for (int idx = tid; idx < 208 * KC; idx += 256) {
        const int r = idx / KC, c = idx % KC;
        float v = 0.f;
        if (r < LPATCH) v = Bptr[(size_t)r * DIM + kc0 + c];
        ldsB[r][c] = v;
      }
      __syncthreads();

      #pragma unroll
      for (int s = 0; s < KC / 4; ++s) {
        const int k0 = 4 * s + 2 * half;
        v2f a;  a.x  = ldsA[l16][k0];              a.y  = ldsA[l16][k0 + 1];
        v2f b0; b0.x = ldsB[t0  * 16 + l16][k0];   b0.y = ldsB[t0  * 16 + l16][k0 + 1];
        v2f b1; b1.x = ldsB[t1c * 16 + l16][k0];   b1.y = ldsB[t1c * 16 + l16][k0 + 1];
        acc0 = __builtin_amdgcn_wmma_f32_16x16x4_f32(false, a, false, b0,
                                                     (short)0, acc0, false, false);
        acc1 = __builtin_amdgcn_wmma_f32_16x16x4_f32(false, a, false, b1,
                                                     (short)0, acc1, false, false);
      }
    }

    // scale by bank inverse norms, fold into running row max
    const float* binvk = binv + ((size_t)(n * BATCH + b) * KTOP + kb) * LPATCH;
    {
      const int col0 = t0 * 16 + l16;
      if (col0 < LPATCH) {
        const float bi = binvk[col0];
        #pragma unroll
        for (int i = 0; i < 8; ++i) mx[i] = fmaxf(mx[i], acc0[i] * bi);
      }
      const int col1 = t1 * 16 + l16;
      if (t1 < 13 && col1 < LPATCH) {
        const float bi = binvk[col1];
        #pragma unroll
        for (int i = 0; i < 8; ++i) mx[i] = fmaxf(mx[i], acc1[i] * bi);
      }
    }
  }

  // reduce the 16 columns held in each 16-lane group (row = i + 8*half)
  #pragma unroll
  for (int m = 8; m >= 1; m >>= 1) {
    #pragma unroll
    for (int i = 0; i < 8; ++i) mx[i] = fmaxf(mx[i], __shfl_xor(mx[i], m, 32));
  }
  if (l16 == 0) {
    #pragma unroll
    for (int i = 0; i < 8; ++i) swmax[wave][half * 8 + i] = mx[i];
  }
  __syncthreads();
  if (tid < 16) {
    float v = swmax[0][tid];
    #pragma unroll
    for (int w = 1; w < 8; ++w) v = fmaxf(v, swmax[w][tid]);
    const int rg = rowbase + tid;
    if (rg < LPATCH) maxsim[arow0 + rg] = v * qinv[arow0 + rg];
  }
}

// ---------------------------------------------------------------------------
// Kernel 5: patch_scores = mean_n (1 - maxsim); anomaly_score / is_unknown
// out layout: [score(8) | unknown(8) | amap(8*224*224) | patch_scores(8*196)]
// ---------------------------------------------------------------------------
#define OUT_AMAP   16
#define OUT_PS     (16 + 8 * 224 * 224)

__global__ __launch_bounds__(256) void k_scores(const float* __restrict__ maxsim,
                                                float* __restrict__ out) {
  const int b = blockIdx.x;
  const int tid = threadIdx.x;
  __shared__ float sred[256];
  float ps = -INFINITY;
  if (tid < LPATCH) {
    float s = 0.f;
    #pragma unroll
    for (int n = 0; n < NL; ++n) s += 1.0f - maxsim[(size_t)(n * BATCH + b) * LPATCH + tid];
    const float val = 0.25f * s;
    out[OUT_PS + b * LPATCH + tid] = val;
    ps = val;
  }
  sred[tid] = ps;
  __syncthreads();
  for (int str = 128; str >= 1; str >>= 1) {
    if (tid < str) sred[tid] = fmaxf(sred[tid], sred[tid + str]);
    __syncthreads();
  }
  if (tid == 0) {
    const float as = sred[0];
    out[b] = as;
    const float z = (as - 0.5f) / 0.1f;
    out[8 + b] = (z > 2.0f) ? 1.0f : 0.0f;
  }
}

// ---------------------------------------------------------------------------
// Kernel 6: 5x5 cross-correlation (zero pad 2) on 14x14, then bilinear 16x
// upsample (half-pixel coords, edge clamp) to 224x224.
// ---------------------------------------------------------------------------
__global__ __launch_bounds__(256) void k_conv_resize(float* __restrict__ out) {
  const int b = blockIdx.x;
  const int tid = threadIdx.x;
  __shared__ float sin_[LPATCH];
  __shared__ float sc[LPATCH];
  __shared__ float gw[25];

  if (tid == 0) {
    float sum = 0.f;
    for (int i = 0; i < 5; ++i)
      for (int j = 0; j < 5; ++j) {
        const float v = expf(-(float)(i * j) * 0.5f);
        gw[i * 5 + j] = v; sum += v;
      }
    const float inv = 1.0f / sum;
    for (int i = 0; i < 25; ++i) gw[i] *= inv;
  }
  if (tid < LPATCH) sin_[tid] = out[OUT_PS + b * LPATCH + tid];
  __syncthreads();

  if (tid < LPATCH) {
    const int y = tid / 14, x = tid % 14;
    float acc = 0.f;
    #pragma unroll
    for (int dy = 0; dy < 5; ++dy) {
      const int iy = y + dy - 2;
      if (iy < 0 || iy >= 14) continue;
      #pragma unroll
      for (int dx = 0; dx < 5; ++dx) {
        const int ix = x + dx - 2;
        if (ix < 0 || ix >= 14) continue;
        acc += sin_[iy * 14 + ix] * gw[dy * 5 + dx];
      }
    }
    sc[tid] = acc;
  }
  __syncthreads();

  for (int p = tid; p < 224 * 224; p += 256) {
    const int oy = p / 224, ox = p % 224;
    const float fy = (oy + 0.5f) * 0.0625f - 0.5f;
    const float fx = (ox + 0.5f) * 0.0625f - 0.5f;
    const float y0f = floorf(fy), x0f = floorf(fx);
    const float wy = fy - y0f, wx = fx - x0f;
    int y0 = (int)y0f, x0 = (int)x0f;
    int y1 = y0 + 1, x1 = x0 + 1;
    y0 = min(max(y0, 0), 13); y1 = min(max(y1, 0), 13);
    x0 = min(max(x0, 0), 13); x1 = min(max(x1, 0), 13);
    const float v =
        (1.f - wy) * ((1.f - wx) * sc[y0 * 14 + x0] + wx * sc[y0 * 14 + x1]) +
        wy         * ((1.f - wx) * sc[y1 * 14 + x0] + wx * sc[y1 * 14 + x1]);
    out[OUT_AMAP + ((size_t)b * 224 + oy) * 224 + ox] = v;
  }
}

// ---------------------------------------------------------------------------
extern "C" void kernel_launch(void* const* d_in, const int* in_sizes, int n_in,
                              void* d_out, int out_size, void* d_ws, size_t ws_size,
                              hipStream_t stream) {
  const float* pf = (const float*)d_in[0];   // patch_feats (4,8,196,768)
  const float* cf = (const float*)d_in[1];   // cls_feats   (4,8,768)
  const float* pb = (const float*)d_in[2];   // patch_banks (4,800,196,768)
  const float* cb = (const float*)d_in[3];   // cls_banks   (4,800,768)
  float* out = (float*)d_out;
  float* ws  = (float*)d_ws;

  int*   topk   = (int*)ws;                  // 40 ints (padded to 64 floats)
  float* qinv   = ws + 64;                   // 6272
  float* binv   = qinv + NL * BATCH * LPATCH;            // 31360
  float* maxsim = binv + NL * BATCH * KTOP * LPATCH;     // 6272

  k_cls_topk<<<BATCH, 256, 0, stream>>>(cf, cb, topk);
  k_rownorm_q<<<(NL * BATCH * LPATCH + 7) / 8, 256, 0, stream>>>(pf, qinv, NL * BATCH * LPATCH);
  k_rownorm_bank<<<(NL * BATCH * KTOP * LPATCH + 7) / 8, 256, 0, stream>>>(pb, topk, binv);

  dim3 grid(13, BATCH, NL);
  k_gemm<<<grid, 256, 0, stream>>>(pf, pb, topk, qinv, binv, maxsim);

  k_scores<<<BATCH, 256, 0, stream>>>(maxsim, out);
  k_conv_resize<<<BATCH, 256, 0, stream>>>(out);
}